// Attn_48206712930921
// MI455X (gfx1250) — compile-verified
//
#include <hip/hip_runtime.h>
#include <hip/hip_bf16.h>

constexpr int kH = 1024;
constexpr int kS = 2048;
constexpr int kB = 64;
#define NEGVAL (-1.0e10f)

typedef __attribute__((ext_vector_type(2))) float v2f;
typedef __attribute__((ext_vector_type(4))) float v4f;
typedef __attribute__((ext_vector_type(8))) float v8f;

// ---------------------------------------------------------------------------
// Stage 0: fold W, weight_vec, bias ->  v = W^T * wvec  (kH floats),
//                                       c = wvec . bias (1 float)
// One wave per 16 output columns; v_wmma_f32_16x16x4_f32 accumulation.
// A-frag (16x4 f32): VGPR0 = K{0,2} halves, VGPR1 = K{1,3};  M = lane%16.
// B-frag: wvec chunk broadcast across all 16 N columns -> every column of D
// equals the per-row dot product, so lanes 0 and 16 hold the 16 results.
// ---------------------------------------------------------------------------
__global__ void __launch_bounds__(32) attn_fold_kernel(
    const float* __restrict__ W, const float* __restrict__ wvec,
    const float* __restrict__ bias, float* __restrict__ v,
    float* __restrict__ cval)
{
    const int lane = threadIdx.x;            // wave32
    const int h0   = blockIdx.x * 16;
    const int m    = lane & 15;
    const int ksel = (lane < 16) ? 0 : 2;

    v8f acc = {0.f, 0.f, 0.f, 0.f, 0.f, 0.f, 0.f, 0.f};
    for (int k0 = 0; k0 < kH; k0 += 4) {
        v2f a, b;
        a.x = W[(size_t)(k0 + ksel)     * kH + h0 + m];   // W^T[m][k] = W[k][h0+m]
        a.y = W[(size_t)(k0 + ksel + 1) * kH + h0 + m];
        b.x = wvec[k0 + ksel];                            // broadcast over N
        b.y = wvec[k0 + ksel + 1];
        acc = __builtin_amdgcn_wmma_f32_16x16x4_f32(
            /*neg_a=*/false, a, /*neg_b=*/false, b,
            /*c_mod=*/(short)0, acc, /*reuse_a=*/false, /*reuse_b=*/false);
    }
    // C/D layout: VGPR r, lane 0 -> (M=r, N=0); lane 16 -> (M=r+8, N=0)
    if (lane == 0) {
        #pragma unroll
        for (int r = 0; r < 8; ++r) v[h0 + r] = acc[r];
    } else if (lane == 16) {
        #pragma unroll
        for (int r = 0; r < 8; ++r) v[h0 + 8 + r] = acc[r];
    }
    if (blockIdx.x == 0) {
        float cs = 0.f;
        for (int i = lane; i < kH; i += 32) cs += wvec[i] * bias[i];
        #pragma unroll
        for (int off = 16; off; off >>= 1) cs += __shfl_xor(cs, off, 32);
        if (lane == 0) cval[0] = cs;
    }
}

// ---------------------------------------------------------------------------
// Stage 1: energies[b,s] = outputs[b,s,:] . v + c, masked by text_lens.
// One wave per row; coalesced 128-bit non-temporal loads (read-once stream,
// keep it out of L2), v staged in LDS, wave32 shuffle reduction.
// ---------------------------------------------------------------------------
__global__ void __launch_bounds__(256) attn_energy_kernel(
    const float* __restrict__ outputs, const int* __restrict__ lens,
    const float* __restrict__ v, const float* __restrict__ cval,
    float* __restrict__ energ)
{
    __shared__ __align__(16) float sv[kH];
    const int t = threadIdx.x;
    #pragma unroll
    for (int i = t; i < kH; i += 256) sv[i] = v[i];
    __syncthreads();

    const int wave = t >> 5, lane = t & 31;
    const long long gidx = (long long)blockIdx.x * 8 + wave;   // row in [0, B*S)
    const int b = (int)(gidx >> 11);                           // / kS
    const int s = (int)(gidx & (kS - 1));
    const float* __restrict__ row = outputs + (size_t)gidx * kH;

    float acc = 0.f;
    #pragma unroll
    for (int j = 0; j < 8; ++j) {
        const int base = (lane + 32 * j) * 4;                  // coalesced float4
        const v4f x  = __builtin_nontemporal_load((const v4f*)(row + base));
        const v4f vv = *(const v4f*)(sv + base);
        acc = fmaf(x.x, vv.x, acc);
        acc = fmaf(x.y, vv.y, acc);
        acc = fmaf(x.z, vv.z, acc);
        acc = fmaf(x.w, vv.w, acc);
    }
    #pragma unroll
    for (int off = 16; off; off >>= 1) acc += __shfl_xor(acc, off, 32);
    if (lane == 0) {
        const float e = acc + cval[0];
        energ[gidx] = (s < lens[b]) ? e : NEGVAL;
    }
}

// ---------------------------------------------------------------------------
// Stage 2: row softmax over S=2048, one 256-thread block (8 waves) per batch.
// Masked entries carry -1e10 -> exp underflows to exactly 0, matching ref.
// ---------------------------------------------------------------------------
__global__ void __launch_bounds__(256) attn_softmax_kernel(
    const float* __restrict__ energ, float* __restrict__ out)
{
    __shared__ float sred[8];
    const int t = threadIdx.x;
    const int wave = t >> 5, lane = t & 31;
    const float* __restrict__ e = energ + (size_t)blockIdx.x * kS;
    float* __restrict__ o = out + (size_t)blockIdx.x * kS;

    float vals[8];
    float lmax = -3.402823466e38f;
    #pragma unroll
    for (int j = 0; j < 8; ++j) {
        vals[j] = e[t + 256 * j];
        lmax = fmaxf(lmax, vals[j]);
    }
    #pragma unroll
    for (int off = 16; off; off >>= 1) lmax = fmaxf(lmax, __shfl_xor(lmax, off, 32));
    if (lane == 0) sred[wave] = lmax;
    __syncthreads();
    float bmax = sred[0];
    #pragma unroll
    for (int w = 1; w < 8; ++w) bmax = fmaxf(bmax, sred[w]);
    __syncthreads();

    float lsum = 0.f;
    #pragma unroll
    for (int j = 0; j < 8; ++j) {
        vals[j] = __expf(vals[j] - bmax);
        lsum += vals[j];
    }
    #pragma unroll
    for (int off = 16; off; off >>= 1) lsum += __shfl_xor(lsum, off, 32);
    if (lane == 0) sred[wave] = lsum;
    __syncthreads();
    float bsum = 0.f;
    #pragma unroll
    for (int w = 0; w < 8; ++w) bsum += sred[w];
    const float inv = 1.0f / bsum;
    #pragma unroll
    for (int j = 0; j < 8; ++j) o[t + 256 * j] = vals[j] * inv;
}

// ---------------------------------------------------------------------------
extern "C" void kernel_launch(void* const* d_in, const int* in_sizes, int n_in,
                              void* d_out, int out_size, void* d_ws, size_t ws_size,
                              hipStream_t stream)
{
    const float* outputs = (const float*)d_in[0];   // [B,S,H] f32
    const int*   lens    = (const int*)  d_in[1];   // [B]
    const float* W       = (const float*)d_in[2];   // [H,H]  f32
    const float* bias    = (const float*)d_in[3];   // [H]    f32
    const float* wvec    = (const float*)d_in[4];   // [H]    f32
    float* out = (float*)d_out;                     // [B,S]  f32

    float* ws    = (float*)d_ws;
    float* v     = ws;                 // kH floats
    float* cval  = ws + kH;            // 1 float
    float* energ = ws + kH + 16;       // B*S floats, 64B-aligned offset

    attn_fold_kernel<<<kH / 16, 32, 0, stream>>>(W, wvec, bias, v, cval);
    attn_energy_kernel<<<(kB * kS) / 8, 256, 0, stream>>>(outputs, lens, v, cval, energ);
    attn_softmax_kernel<<<kB, 256, 0, stream>>>(energ, out);

    (void)in_sizes; (void)n_in; (void)out_size; (void)ws_size;
}